// DecomposedKLDAddLoss_47510928228666
// MI455X (gfx1250) — compile-verified
//
#include <hip/hip_runtime.h>
#include <math.h>

// ---------------------------------------------------------------------------
// DecomposedKLDAddLoss on MI455X (gfx1250, wave32)
//   M[i,j] = sum_d log N(z[i,d]; loc[j,d], scale[j,d])
//          = c_j + sum_d (beta_jd * z_id + gamma_jd * z_id^2)
//   -> exact f32 GEMM (K=128) via V_WMMA_F32_16X16X4_F32
//   per-dim logsumexp_j -> online LSE with LDS-staged coefficient tiles
// ---------------------------------------------------------------------------

typedef __attribute__((ext_vector_type(2))) float v2f;
typedef __attribute__((ext_vector_type(8))) float v8f;

#define B_N 2048
#define D_N 64
#define K_N 128                      // interleaved (z, z^2) / (beta, gamma)
#define LOG2PI_F 1.8378770664093453f
#define NORMC 18.4443972f            // log(B * DATA_SIZE) = log(2048*50000)
#define NEG_INF_F (-3.0e38f)

// workspace layout (float offsets)
#define AZ_OFF  0                            // [B_N][K_N]  (z, z^2) interleaved
#define BC_OFF  (AZ_OFF + B_N * K_N)         // [B_N][K_N]  (beta, gamma) interleaved
#define AL_OFF  (BC_OFF + B_N * K_N)         // [B_N][D_N]  alpha
#define CJ_OFF  (AL_OFF + B_N * D_N)         // [B_N]       c_j = sum_d alpha
#define M_OFF   (CJ_OFF + B_N)               // [B_N][B_N]  pairwise joint log-density
#define LQ_OFF  (M_OFF + B_N * B_N)          // [B_N]       raw row LSE of M
#define LQP_OFF (LQ_OFF + B_N)               // [B_N]       log_qz_prod (norm applied)

// ---------------------------------------------------------------------------
// Kernel 1: per-element coefficients + per-row constant c_j
// grid = B_N blocks, block = D_N (=64) threads
// ---------------------------------------------------------------------------
__global__ void __launch_bounds__(64) prep_kernel(const float* __restrict__ z,
                                                  const float* __restrict__ loc,
                                                  const float* __restrict__ scale,
                                                  float* __restrict__ ws) {
  float* Az = ws + AZ_OFF;
  float* Bc = ws + BC_OFF;
  float* Al = ws + AL_OFF;
  float* Cj = ws + CJ_OFF;

  const int row = blockIdx.x;
  const int d   = threadIdx.x;
  const int idx = row * D_N + d;

  const float zv = z[idx];
  const float lv = loc[idx];
  const float sv = scale[idx];
  const float inv2  = 1.0f / (sv * sv);
  const float beta  = lv * inv2;
  const float gamma = -0.5f * inv2;
  const float alpha = -0.5f * LOG2PI_F - __logf(sv) - 0.5f * lv * lv * inv2;

  Az[row * K_N + 2 * d]     = zv;
  Az[row * K_N + 2 * d + 1] = zv * zv;
  Bc[row * K_N + 2 * d]     = beta;
  Bc[row * K_N + 2 * d + 1] = gamma;
  Al[idx] = alpha;

  __shared__ float red[D_N];
  red[d] = alpha;
  __syncthreads();
  for (int s = D_N / 2; s > 0; s >>= 1) {
    if (d < s) red[d] += red[d + s];
    __syncthreads();
  }
  if (d == 0) Cj[row] = red[0];
}

// ---------------------------------------------------------------------------
// Kernel 2: pairwise joint log-density GEMM via V_WMMA_F32_16X16X4_F32
// one 16x16 output tile per wave32; K = 128 -> 32 WMMA steps
// grid = (128*128 tiles)/8 waves = 2048 blocks, block = 256 threads (8 waves)
// ---------------------------------------------------------------------------
__global__ void __launch_bounds__(256) pair_gemm_kernel(float* __restrict__ ws) {
  const float* Az = ws + AZ_OFF;
  const float* Bc = ws + BC_OFF;
  const float* Cj = ws + CJ_OFF;
  float*       M  = ws + M_OFF;

  const int wave = (blockIdx.x * blockDim.x + threadIdx.x) >> 5;  // 0..16383
  const int lane = threadIdx.x & 31;
  const int ti = wave >> 7;            // 0..127
  const int tj = wave & 127;           // 0..127
  const int i0 = ti * 16;
  const int j0 = tj * 16;

  // ISA 7.12.2 wave32 fragment layout for 16x16x4 f32:
  //   A: lanes 0-15 hold (K=kk,kk+1) of row M=lane; lanes 16-31 hold (kk+2,kk+3)
  //   B: mirrored for columns N
  const int half = lane >> 4;          // 0 or 1
  const int l16  = lane & 15;
  const float* arow = Az + (size_t)(i0 + l16) * K_N + 2 * half;
  const float* brow = Bc + (size_t)(j0 + l16) * K_N + 2 * half;

  v8f acc = {0.f, 0.f, 0.f, 0.f, 0.f, 0.f, 0.f, 0.f};

#pragma unroll
  for (int kk = 0; kk < K_N; kk += 4) {
    const float2 av = *reinterpret_cast<const float2*>(arow + kk);
    const float2 bv = *reinterpret_cast<const float2*>(brow + kk);
    v2f a, b;
    a[0] = av.x; a[1] = av.y;
    b[0] = bv.x; b[1] = bv.y;
    acc = __builtin_amdgcn_wmma_f32_16x16x4_f32(
        /*neg_a=*/false, a, /*neg_b=*/false, b,
        /*c_mod=*/(short)0, acc, /*reuse_a=*/false, /*reuse_b=*/false);
  }

  const float cadd = Cj[j0 + l16];     // per-column constant c_j
#pragma unroll
  for (int r = 0; r < 8; r++) {
    const int row = i0 + r + 8 * half; // C/D layout: VGPR r -> row r / r+8
    const int col = j0 + l16;
    M[(size_t)row * B_N + col] = acc[r] + cadd;
  }
}

// ---------------------------------------------------------------------------
// Kernel 3: row-wise logsumexp of M  ->  lq_raw[i]  (norm applied later)
// grid = B_N blocks, block = 256 threads
// ---------------------------------------------------------------------------
__global__ void __launch_bounds__(256) row_lse_kernel(float* __restrict__ ws) {
  const float* M   = ws + M_OFF;
  float*       lqr = ws + LQ_OFF;
  const int row = blockIdx.x;
  const int t   = threadIdx.x;

  float m = NEG_INF_F, s = 0.f;
  for (int c = t; c < B_N; c += 256) {
    const float x = M[(size_t)row * B_N + c];
    const float mo = m;
    m = fmaxf(m, x);
    s = s * __expf(mo - m) + __expf(x - m);
  }
  __shared__ float sm[256], ss[256];
  sm[t] = m; ss[t] = s;
  __syncthreads();
  for (int st = 128; st > 0; st >>= 1) {
    if (t < st) {
      const float m1 = sm[t], s1 = ss[t];
      const float m2 = sm[t + st], s2 = ss[t + st];
      const float mm = fmaxf(m1, m2);
      sm[t] = mm;
      ss[t] = s1 * __expf(m1 - mm) + s2 * __expf(m2 - mm);
    }
    __syncthreads();
  }
  if (t == 0) lqr[row] = sm[0] + __logf(ss[0]);
}

// ---------------------------------------------------------------------------
// Kernel 4: per-dimension online logsumexp over j -> log_qz_prod[i]
// block owns 32 rows i, all 64 dims; j-tiles of 64 staged in LDS (48 KB)
// thread t: d = t&63, handles i_local = (t>>6) + 4*p, p=0..7
// grid = B_N/32 = 64 blocks, block = 256 threads
// ---------------------------------------------------------------------------
#define ITILE 32
__global__ void __launch_bounds__(256) perdim_lse_kernel(float* __restrict__ ws) {
  const float* Az  = ws + AZ_OFF;
  const float* Bc  = ws + BC_OFF;
  const float* Al  = ws + AL_OFF;
  float*       lqp = ws + LQP_OFF;

  __shared__ float lA[64 * 64];
  __shared__ float lB[64 * 64];
  __shared__ float lG[64 * 64];
  __shared__ float acc_i[ITILE];

  const int t  = threadIdx.x;
  const int d  = t & 63;
  const int iq = t >> 6;               // 0..3
  const int i0 = blockIdx.x * ITILE;

  float zv[8], z2[8], m[8], s[8];
#pragma unroll
  for (int p = 0; p < 8; p++) {
    const int i = i0 + iq + 4 * p;
    const float2 az = *reinterpret_cast<const float2*>(Az + (size_t)i * K_N + 2 * d);
    zv[p] = az.x; z2[p] = az.y;
    m[p] = NEG_INF_F; s[p] = 0.f;
  }
  if (t < ITILE) acc_i[t] = 0.f;

  for (int jb = 0; jb < B_N; jb += 64) {
    __syncthreads();
    for (int idx = t; idx < 64 * 64; idx += 256) {
      const int j  = jb + (idx >> 6);
      const int dd = idx & 63;
      lA[idx] = Al[(size_t)j * D_N + dd];
      const float2 bg = *reinterpret_cast<const float2*>(Bc + (size_t)j * K_N + 2 * dd);
      lB[idx] = bg.x;
      lG[idx] = bg.y;
    }
    __syncthreads();

    for (int jj = 0; jj < 64; jj++) {
      const float a = lA[jj * 64 + d];
      const float b = lB[jj * 64 + d];
      const float g = lG[jj * 64 + d];
#pragma unroll
      for (int p = 0; p < 8; p++) {
        const float x  = fmaf(g, z2[p], fmaf(b, zv[p], a));
        const float mo = m[p];
        m[p] = fmaxf(m[p], x);
        s[p] = s[p] * __expf(mo - m[p]) + __expf(x - m[p]);
      }
    }
  }

  __syncthreads();
#pragma unroll
  for (int p = 0; p < 8; p++) {
    const float lse = m[p] + __logf(s[p]);
    atomicAdd(&acc_i[iq + 4 * p], lse - NORMC);  // ds_add_f32
  }
  __syncthreads();
  if (t < ITILE) lqp[i0 + t] = acc_i[t];
}

// ---------------------------------------------------------------------------
// Kernel 5: final loss = mean(mi) + mean(tc) + mean(kl)
// lqcx[i] = M[i,i] (diagonal of pairwise matrix), lpz from z directly
// ---------------------------------------------------------------------------
__global__ void __launch_bounds__(256) finalize_kernel(const float* __restrict__ zin,
                                                       const float* __restrict__ ws,
                                                       float* __restrict__ out) {
  const float* M   = ws + M_OFF;
  const float* lqr = ws + LQ_OFF;
  const float* lqp = ws + LQP_OFF;
  const int t = threadIdx.x;

  float mi = 0.f, tc = 0.f, kl = 0.f;
  for (int i = t; i < B_N; i += 256) {
    const float lqcx = M[(size_t)i * B_N + i];
    const float lq   = lqr[i] - NORMC;
    const float lp   = lqp[i];
    float sz2 = 0.f;
    const float* zr = zin + (size_t)i * D_N;
    for (int dd = 0; dd < D_N; dd++) {
      const float zz = zr[dd];
      sz2 = fmaf(zz, zz, sz2);
    }
    const float lpz = -0.5f * LOG2PI_F * (float)D_N - 0.5f * sz2;
    mi += lqcx - lq;
    tc += lq - lp;
    kl += lp - lpz;
  }
  __shared__ float smi[256], stc[256], skl[256];
  smi[t] = mi; stc[t] = tc; skl[t] = kl;
  __syncthreads();
  for (int st = 128; st > 0; st >>= 1) {
    if (t < st) {
      smi[t] += smi[t + st];
      stc[t] += stc[t + st];
      skl[t] += skl[t + st];
    }
    __syncthreads();
  }
  if (t == 0) {
    const float invB = 1.0f / (float)B_N;
    out[0] = (smi[0] * invB) + (stc[0] * invB) + (skl[0] * invB);
  }
}

// ---------------------------------------------------------------------------
extern "C" void kernel_launch(void* const* d_in, const int* in_sizes, int n_in,
                              void* d_out, int out_size, void* d_ws, size_t ws_size,
                              hipStream_t stream) {
  (void)in_sizes; (void)n_in; (void)out_size; (void)ws_size;
  const float* z     = (const float*)d_in[0];
  const float* loc   = (const float*)d_in[1];
  const float* scale = (const float*)d_in[2];
  float* ws  = (float*)d_ws;
  float* out = (float*)d_out;

  prep_kernel<<<B_N, D_N, 0, stream>>>(z, loc, scale, ws);
  pair_gemm_kernel<<<(128 * 128) / 8, 256, 0, stream>>>(ws);   // 8 waves/block, 1 tile/wave
  row_lse_kernel<<<B_N, 256, 0, stream>>>(ws);
  perdim_lse_kernel<<<B_N / ITILE, 256, 0, stream>>>(ws);
  finalize_kernel<<<1, 256, 0, stream>>>(z, ws, out);
}